// RecurrentLinearAttention_4045859193551
// MI455X (gfx1250) — compile-verified
//
#include <hip/hip_runtime.h>

// CDNA5 / gfx1250: wave32, WMMA f32 16x16x4
typedef __attribute__((ext_vector_type(4))) float v4f;
typedef __attribute__((ext_vector_type(2))) float v2f;
typedef __attribute__((ext_vector_type(8))) float v8f;

// Problem constants from the reference: N=1024, H=16, D=64, M=64
#define NHPAIRS 16384     // N*H
#define DDIM 64
#define MDIM 64
// d_out layout: [V (NH*M)] [Si_new (NH*D*M)] [Zi_new (NH*D)]
constexpr int OFF_S = NHPAIRS * MDIM;                  // 1,048,576
constexpr int OFF_Z = OFF_S + NHPAIRS * DDIM * MDIM;   // 68,157,440

__device__ __forceinline__ float elu_p1(float x) {
  // elu(x) + 1 == (x > 0) ? x + 1 : exp(x)
  return x > 0.0f ? x + 1.0f : __expf(x);
}

__global__ __launch_bounds__(256) void rla_step_kernel(
    const float* __restrict__ q, const float* __restrict__ k,
    const float* __restrict__ v, const float* __restrict__ Si,
    const float* __restrict__ Zi, float* __restrict__ out)
{
  __shared__ __align__(16) float sS[DDIM * MDIM];   // staged Si_new (16 KB)
  __shared__ float sQ[DDIM];
  __shared__ float sK[DDIM];
  __shared__ __align__(16) float sV[MDIM];
  __shared__ float zsum;

  const int t    = threadIdx.x;
  const int b    = blockIdx.x;          // (n,h) pair index
  const int lane = t & 31;
  const int wave = t >> 5;

  if (t == 0) zsum = 0.0f;

  // ---- Phase 1: feature maps, Zi_new, partial Q.Zi_new ----
  float p = 0.0f;
  if (t < 64) {
    const float Qf = elu_p1(q[b * 64 + t]);
    const float Kf = elu_p1(k[b * 64 + t]);
    const float vv = v[b * 64 + t];
    const float zn = Zi[b * 64 + t] + Kf;
    out[OFF_Z + b * 64 + t] = zn;       // Zi_new
    sQ[t] = Qf; sK[t] = Kf; sV[t] = vv;
    p = Qf * zn;
  }
  __syncthreads();                       // zsum init + sQ/sK/sV visible
  if (t < 64) {                          // waves 0,1 reduce 32 lanes each
    for (int off = 16; off; off >>= 1) p += __shfl_xor(p, off, 32);
    if (lane == 0) atomicAdd(&zsum, p);  // ds_add_f32
  }

  // ---- Phase 2: stream Si -> Si_new (rank-1 update), coalesced b128 ----
  // float4 index f = i*256 + t  =>  d = i*16 + t/16, m-quad = (t%16)*4
  const v4f* SiV = (const v4f*)(Si + (size_t)b * 4096);
  v4f*       SoV = (v4f*)(out + OFF_S + (size_t)b * 4096);
  v4f*       sSV = (v4f*)sS;
  const v4f  vv4 = ((const v4f*)sV)[t & 15];
#pragma unroll
  for (int i = 0; i < 4; ++i) {
    const int idx = i * 256 + t;
    const int d   = i * 16 + (t >> 4);
    v4f s = SiV[idx];
    s += sK[d] * vv4;                    // Si + K[d] * value[m..m+3]
    SoV[idx] = s;                        // Si_new to HBM (coalesced 512B/wave)
    sSV[idx] = s;                        // stage for WMMA matvec
  }
  __syncthreads();                       // sS complete, zsum complete

  const float Zs = 1.0f / (zsum + 1e-6f);

  // ---- Phase 3: V = Zs * (Q . Si_new) via chained V_WMMA_F32_16X16X4_F32 ----
  // Waves 0..3 each own a 16-column tile. A row 0 = Q chunk (ISA 16x4 A layout:
  // lane0 -> K=0(v0),K=1(v1); lane16 -> K=2(v0),K=3(v1); other lanes zero).
  // B = 4x16 row chunk of Si_new from LDS. D row 0 accumulates V.
  // Q is loaded unconditionally by all lanes and masked with cndmask selects
  // (not EXEC predication) so the scheduler can pipeline the ds loads freely.
  if (wave < 4) {
    const int m0    = wave * 16;
    const int lhalf = lane >> 4;         // 0: lanes 0-15, 1: lanes 16-31
    const int ln    = lane & 15;
    const bool row0 = (ln == 0);         // lanes 0 and 16 carry the Q row
    v8f c = {};
#pragma unroll
    for (int kk = 0; kk < 64; kk += 4) {
      const float qx = sQ[kk + 2 * lhalf];          // unconditional ds_load
      const float qy = sQ[kk + 2 * lhalf + 1];
      v2f a;
      a.x = row0 ? qx : 0.0f;                       // v_cndmask, no saveexec
      a.y = row0 ? qy : 0.0f;
      v2f bb;
      bb.x = sS[(kk + 2 * lhalf) * 64 + m0 + ln];       // B[K=0 / K=2]
      bb.y = sS[(kk + 2 * lhalf + 1) * 64 + m0 + ln];   // B[K=1 / K=3]
      c = __builtin_amdgcn_wmma_f32_16x16x4_f32(
          false, a, false, bb, (short)0, c, false, false);
    }
    if (lane < 16) {                     // D[M=0][N=lane] lives in c[0], lanes 0-15
      out[b * 64 + m0 + lane] = c[0] * Zs;
    }
  }
}

extern "C" void kernel_launch(void* const* d_in, const int* in_sizes, int n_in,
                              void* d_out, int out_size, void* d_ws, size_t ws_size,
                              hipStream_t stream) {
  (void)in_sizes; (void)n_in; (void)out_size; (void)d_ws; (void)ws_size;
  const float* q  = (const float*)d_in[0];
  const float* k  = (const float*)d_in[1];
  const float* v  = (const float*)d_in[2];
  const float* Si = (const float*)d_in[3];
  const float* Zi = (const float*)d_in[4];
  float* out = (float*)d_out;
  rla_step_kernel<<<dim3(NHPAIRS), dim3(256), 0, stream>>>(q, k, v, Si, Zi, out);
}